// NodeAttention_79147657330881
// MI455X (gfx1250) — compile-verified
//
#include <hip/hip_runtime.h>
#include <hip/hip_bf16.h>
#include <stdint.h>

#define N_NODES   100000
#define N_EDGES   3200000
#define NODE_DIM  256
#define ATTN_DIM  32
#define N_TILES   (N_NODES / 16)      // 6250, exact

typedef __attribute__((ext_vector_type(2))) float v2f;
typedef __attribute__((ext_vector_type(8))) float v8f;

// ---------------------------------------------------------------------------
// init: seg_max = -inf, seg_sum = 0
// ---------------------------------------------------------------------------
__global__ void seg_init_kernel(float* __restrict__ segmax, float* __restrict__ segsum) {
    int i = blockIdx.x * blockDim.x + threadIdx.x;
    if (i < N_NODES) {
        segmax[i] = -__builtin_inff();
        segsum[i] = 0.0f;
    }
}

// ---------------------------------------------------------------------------
// Projection: q = x@Wq + bq, k = x@Wk + bk via V_WMMA_F32_16X16X4_F32.
// Also fuses the x -> d_out copy (output tuple element 0).
//
// LDS holds Wq (cols 0..31) and Wk (cols 32..63) transposed:
//   word(n, k) = n*256 + (k ^ (4*(n&15)))
// XOR swizzle keeps float2 (8B) alignment (xor value is a multiple of 4) and
// makes the 16 lanes of each B-load hit 16 distinct banks; lanes 16-31 read
// k+2 -> banks differing mod 4 from lanes 0-15 -> fully conflict-free b64 DS.
// Cols n and n+16/n+32/n+48 share the same swizzle (4*16 == 64 == 0 mod 64).
// Total LDS: 64 cols * 256 * 4B = 64 KB exactly.
//
// WMMA f32 16x16x4 per-lane layout (ISA 7.12.2):
//   A: lane L holds x[row = base + (L&15)][k0], x[..][k0+1], k0 = kb + 2*(L>>4)
//   B: lane L holds W[k0][n], W[k0+1][n],       n  = (L&15) + 16*ntile
//   C/D: VGPR r of lane L = out[row = base + r + 8*(L>>4)][n]
// ---------------------------------------------------------------------------
__global__ __launch_bounds__(256) void proj_kernel(
    const float* __restrict__ x,
    const float* __restrict__ Wq, const float* __restrict__ bq,
    const float* __restrict__ Wk, const float* __restrict__ bk,
    float* __restrict__ qOut, float* __restrict__ kOut,
    float* __restrict__ xOut)
{
    __shared__ float ldsW[64 * 256];   // 64 KB

    // ---- stage Wq/Wk transposed+swizzled into LDS (coalesced global reads)
    for (int idx = threadIdx.x; idx < NODE_DIM * ATTN_DIM; idx += 256) {
        int kk = idx >> 5;          // 0..255
        int n  = idx & 31;          // 0..31
        int sw = (n << 2) & 63;
        ldsW[n * 256        + (kk ^ sw)] = Wq[idx];
        ldsW[(n + 32) * 256 + (kk ^ sw)] = Wk[idx];
    }
    __syncthreads();

    const int wave = threadIdx.x >> 5;
    const int lane = threadIdx.x & 31;
    const int m    = lane & 15;
    const int hi   = lane >> 4;        // 0 or 1
    const int koff = hi * 2;           // A/B K sub-offset

    int tile = blockIdx.x * 8 + wave;
    if (tile >= N_TILES) return;       // whole-wave uniform; EXEC stays all-1s

    const int rowBase = tile * 16;
    const float* xRow  = x    + (size_t)(rowBase + m) * NODE_DIM;
    float*       xoRow = xOut + (size_t)(rowBase + m) * NODE_DIM;

    // bias-initialized accumulators (bias depends only on column n)
    v8f cq0, cq1, ck0, ck1;
    {
        float b0 = bq[m], b1 = bq[16 + m], b2 = bk[m], b3 = bk[16 + m];
#pragma unroll
        for (int r = 0; r < 8; ++r) { cq0[r] = b0; cq1[r] = b1; ck0[r] = b2; ck1[r] = b3; }
    }

    const int   sw   = (m << 2) & 63;          // shared by all 4 B columns
    const float* bQ0 = ldsW + (size_t)( m      ) * 256;
    const float* bQ1 = ldsW + (size_t)(16 + m  ) * 256;
    const float* bK0 = ldsW + (size_t)(32 + m  ) * 256;
    const float* bK1 = ldsW + (size_t)(48 + m  ) * 256;

    for (int kb = 0; kb < NODE_DIM; kb += 4) {
        v2f a = *(const v2f*)(xRow + kb + koff);
        *(v2f*)(xoRow + kb + koff) = a;        // fused x -> out copy (each elem once)

        const int kx = (kb + koff) ^ sw;
        v2f b;
        b = *(const v2f*)(bQ0 + kx);
        cq0 = __builtin_amdgcn_wmma_f32_16x16x4_f32(false, a, false, b, (short)0, cq0, false, false);
        b = *(const v2f*)(bQ1 + kx);
        cq1 = __builtin_amdgcn_wmma_f32_16x16x4_f32(false, a, false, b, (short)0, cq1, false, false);
        b = *(const v2f*)(bK0 + kx);
        ck0 = __builtin_amdgcn_wmma_f32_16x16x4_f32(false, a, false, b, (short)0, ck0, false, false);
        b = *(const v2f*)(bK1 + kx);
        ck1 = __builtin_amdgcn_wmma_f32_16x16x4_f32(false, a, false, b, (short)0, ck1, false, false);
    }

    // scatter D: 16-lane coalesced 64B column stores
#pragma unroll
    for (int r = 0; r < 8; ++r) {
        size_t row = (size_t)(rowBase + r + hi * 8);
        qOut[row * ATTN_DIM + m]      = cq0[r];
        qOut[row * ATTN_DIM + 16 + m] = cq1[r];
        kOut[row * ATTN_DIM + m]      = ck0[r];
        kOut[row * ATTN_DIM + 16 + m] = ck1[r];
    }
}

// ---------------------------------------------------------------------------
// float atomic max via signed-max (val>=0) / unsigned-min (val<0) bit trick;
// correct with -inf (0xFF800000) initialization.
// ---------------------------------------------------------------------------
__device__ __forceinline__ void atomicMaxF32(float* addr, float val) {
    if (val >= 0.0f)
        atomicMax((int*)addr, __float_as_int(val));
    else
        atomicMin((unsigned int*)addr, (unsigned int)__float_as_int(val));
}

// ---------------------------------------------------------------------------
// pass 1: score[e] = dot32(q[src], k[dst]); seg_max[src] = max(...)
// one edge per lane; each gather is exactly one 128B cacheline
// ---------------------------------------------------------------------------
__global__ __launch_bounds__(256) void edge_scores_kernel(
    const long long* __restrict__ ei,
    const float* __restrict__ q, const float* __restrict__ k,
    float* __restrict__ scores, float* __restrict__ segmax)
{
    int e = blockIdx.x * blockDim.x + threadIdx.x;
    if (e >= N_EDGES) return;
    long long s = ei[e];
    long long t = ei[N_EDGES + e];
    const float4* qp = (const float4*)(q + (size_t)s * ATTN_DIM);
    const float4* kp = (const float4*)(k + (size_t)t * ATTN_DIM);
    float acc = 0.0f;
#pragma unroll
    for (int i = 0; i < 8; ++i) {
        float4 a = qp[i], b = kp[i];
        acc = fmaf(a.x, b.x, acc);
        acc = fmaf(a.y, b.y, acc);
        acc = fmaf(a.z, b.z, acc);
        acc = fmaf(a.w, b.w, acc);
    }
    scores[e] = acc;
    atomicMaxF32(&segmax[s], acc);
}

// ---------------------------------------------------------------------------
// pass 2: attn[e] = exp(score - seg_max[src]); seg_sum[src] += attn[e]
// ---------------------------------------------------------------------------
__global__ __launch_bounds__(256) void edge_expsum_kernel(
    const long long* __restrict__ ei,
    const float* __restrict__ segmax,
    float* __restrict__ attn, float* __restrict__ segsum)
{
    int e = blockIdx.x * blockDim.x + threadIdx.x;
    if (e >= N_EDGES) return;
    long long s = ei[e];
    float ex = __expf(attn[e] - segmax[s]);
    attn[e] = ex;
    atomicAdd(&segsum[s], ex);
}

// ---------------------------------------------------------------------------
// pass 3: attn[e] /= seg_sum[src]
// ---------------------------------------------------------------------------
__global__ __launch_bounds__(256) void edge_norm_kernel(
    const long long* __restrict__ ei,
    const float* __restrict__ segsum,
    float* __restrict__ attn)
{
    int e = blockIdx.x * blockDim.x + threadIdx.x;
    if (e >= N_EDGES) return;
    long long s = ei[e];
    attn[e] = attn[e] / segsum[s];
}

// ---------------------------------------------------------------------------
extern "C" void kernel_launch(void* const* d_in, const int* in_sizes, int n_in,
                              void* d_out, int out_size, void* d_ws, size_t ws_size,
                              hipStream_t stream) {
    const float*     x  = (const float*)d_in[0];
    const long long* ei = (const long long*)d_in[1];   // int64 [2, E]
    const float*     Wq = (const float*)d_in[2];
    const float*     bq = (const float*)d_in[3];
    const float*     Wk = (const float*)d_in[4];
    const float*     bk = (const float*)d_in[5];

    float* outX = (float*)d_out;                                   // [N, 256]
    float* attn = outX + (size_t)N_NODES * NODE_DIM;               // [E]

    // workspace: q | k | seg_max | seg_sum  (~26.4 MB)
    float* q      = (float*)d_ws;
    float* k      = q + (size_t)N_NODES * ATTN_DIM;
    float* segmax = k + (size_t)N_NODES * ATTN_DIM;
    float* segsum = segmax + N_NODES;

    seg_init_kernel<<<(N_NODES + 255) / 256, 256, 0, stream>>>(segmax, segsum);
    proj_kernel<<<(N_TILES + 7) / 8, 256, 0, stream>>>(x, Wq, bq, Wk, bk, q, k, outX);

    const int eb = (N_EDGES + 255) / 256;
    edge_scores_kernel<<<eb, 256, 0, stream>>>(ei, q, k, attn, segmax);
    edge_expsum_kernel<<<eb, 256, 0, stream>>>(ei, segmax, attn, segsum);
    edge_norm_kernel  <<<eb, 256, 0, stream>>>(ei, segsum, attn);
}